// PDECell_35270271435009
// MI455X (gfx1250) — compile-verified
//
#include <hip/hip_runtime.h>
#include <math.h>

// ---------------------------------------------------------------------------
// PDECell for MI455X (gfx1250, wave32). Dominant cost: 7 Riccati steps of
// dense 1024^3 fp32 GEMMs -> V_WMMA_F32_16X16X4_F32 fed from LDS tiles staged
// via the CDNA5 async copy path (GLOBAL_LOAD_ASYNC_TO_LDS_* + s_wait_asynccnt)
// double-buffered so DMA of tile k+1 overlaps WMMA on tile k. B/A^T tiles are
// stored pair-interleaved in LDS so each WMMA fragment is one aligned
// ds_load_b64 into an even VGPR pair (no repacking movs).
// ---------------------------------------------------------------------------

#define NN 1024
#define EPSV 1e-8f
#define RICCATI_ITERS 6
#define ODE_UNFOLDS 6
#define KC 16            // K-chunk staged per LDS buffer

typedef float v2f __attribute__((ext_vector_type(2)));
typedef float v8f __attribute__((ext_vector_type(8)));

__device__ __forceinline__ void async_b128(unsigned lds_off,
                                           const float* gaddr) {
  asm volatile("global_load_async_to_lds_b128 %0, %1, off"
               :: "v"(lds_off), "v"(gaddr) : "memory");
}
__device__ __forceinline__ void async_b32(unsigned lds_off,
                                          const float* gaddr) {
  asm volatile("global_load_async_to_lds_b32 %0, %1, off"
               :: "v"(lds_off), "v"(gaddr) : "memory");
}
__device__ __forceinline__ unsigned lds_off_of(const void* p) {
  return (unsigned)(unsigned long long)p;  // generic LDS addr low 32b = offset
}

// ---------------------------------------------------------------------------
// WMMA fp32 GEMM:  C = alpha * op(A) @ B + beta * C     (square n x n)
// Block = 128 threads = 4 waves; 64x64 C tile (wave w -> rows [16w,16w+16)).
// Per KC=16 chunk: async-stage A and B into LDS, then 4 k-steps x 4 N-tiles
// of v_wmma_f32_16x16x4_f32. Layouts:
//   B (and A^T) tile: pair-interleaved [k/2][n][k&1], k-pair stride BPST=160
//     words (lane-half bank offset 32 -> conflict-free, b64-aligned pairs).
//   A (no-trans) tile: row-major [m][k], stride 20 words (conflict-free).
// ---------------------------------------------------------------------------
template <int TA>
__global__ __launch_bounds__(128) void gemm_f32_wmma(
    const float* __restrict__ A, const float* __restrict__ B,
    float* __restrict__ C, float alpha, float beta, int n) {
  constexpr int BPST = 160;                     // words per k-pair block
  constexpr int AST  = 20;                      // A (no-trans) row stride
  constexpr int BSZ  = (KC / 2) * BPST;         // 1280 words per buffer
  constexpr int ASZ  = TA ? BSZ : (64 * AST);
  constexpr int NASYNC = TA ? 16 : 10;          // async ops per wave per chunk
  __shared__ float As[2][ASZ];
  __shared__ float Bs[2][BSZ];

  const int tid   = threadIdx.x;
  const int lane  = tid & 31;
  const int wave  = tid >> 5;
  const int row0g = blockIdx.y * 64;
  const int col0g = blockIdx.x * 64;
  const int khalf = (lane >> 4) << 1;           // 0 | 2
  const int idx   = lane & 15;                  // M for A, N for B/C
  const int mloc  = wave * 16 + idx;

  // issue one chunk's async copies (uniform trip counts; per-wave op count =
  // NASYNC, used by s_wait_asynccnt below)
  auto issue = [&](int k0, int buf) {
    // B tile 16x64 -> pair-interleaved scatter (8 x b32 per thread)
    for (int q = tid; q < 1024; q += 128) {
      const int k = q >> 6, nl = q & 63;
      async_b32(lds_off_of(&Bs[buf][(k >> 1) * BPST + nl * 2 + (k & 1)]),
                &B[(size_t)(k0 + k) * n + col0g + nl]);
    }
    if (TA) {  // A^T tile 16x64 (rows=k, cols=m) -> pair-interleaved scatter
      for (int q = tid; q < 1024; q += 128) {
        const int k = q >> 6, ml = q & 63;
        async_b32(lds_off_of(&As[buf][(k >> 1) * BPST + ml * 2 + (k & 1)]),
                  &A[(size_t)(k0 + k) * n + row0g + ml]);
      }
    } else {   // A tile 64x16 row-major, padded stride AST (2 x b128)
      for (int q = tid; q < 256; q += 128) {
        const int row = q >> 2, c4 = q & 3;
        async_b128(lds_off_of(&As[buf][row * AST + c4 * 4]),
                   &A[(size_t)(row0g + row) * n + k0 + c4 * 4]);
      }
    }
  };

  v8f acc[4] = {};
  const int nc = n / KC;
  issue(0, 0);
  for (int c = 0; c < nc; ++c) {
    const int buf = c & 1;
    if (c + 1 < nc) {
      issue((c + 1) * KC, 1 - buf);
      asm volatile("s_wait_asynccnt %0" :: "i"(NASYNC) : "memory");
    } else {
      asm volatile("s_wait_asynccnt 0" ::: "memory");
    }
    __syncthreads();  // all waves' DMA done -> tile visible

#pragma unroll
    for (int kk = 0; kk < KC; kk += 4) {
      const int kp = (kk + khalf) >> 1;   // k-pair index (kk+khalf is even)
      v2f a;
      if (TA) {
        const float2 ap =
            *(const float2*)&As[buf][kp * BPST + mloc * 2];
        a.x = ap.x; a.y = ap.y;
      } else {
        const float2 ap =
            *(const float2*)&As[buf][mloc * AST + kk + khalf];
        a.x = ap.x; a.y = ap.y;
      }
#pragma unroll
      for (int t = 0; t < 4; ++t) {
        const float2 bp =
            *(const float2*)&Bs[buf][kp * BPST + (t * 16 + idx) * 2];
        v2f b; b.x = bp.x; b.y = bp.y;
        acc[t] = __builtin_amdgcn_wmma_f32_16x16x4_f32(
            false, a, false, b, (short)0, acc[t], false, false);
      }
    }
    __syncthreads();  // tile consumed; safe to overwrite next iteration
  }

  const int mbase = (lane >> 4) * 8;
#pragma unroll
  for (int t = 0; t < 4; ++t) {
    const int c0 = col0g + t * 16;
#pragma unroll
    for (int r = 0; r < 8; ++r) {
      float* p = &C[(size_t)(row0g + wave * 16 + mbase + r) * n + c0 + idx];
      float res = alpha * acc[t][r];
      if (beta != 0.0f) res += beta * (*p);
      *p = res;
    }
  }
}

// ---------------------------------------------------------------------------
// In-place Gauss-Jordan solve of M X = R (X left in R). Single 1024-thread
// workgroup, pivot row in LDS, float4 (b128) row updates.
// ---------------------------------------------------------------------------
__global__ __launch_bounds__(1024) void gauss_jordan_solve(
    float* __restrict__ M, float* __restrict__ R, int n) {
  __shared__ float mk[NN];
  __shared__ float rk[NN];
  __shared__ float pivinv;
  const int tid = threadIdx.x;
  const int n4 = n >> 2;
  for (int k = 0; k < n; ++k) {
    if (tid == 0) pivinv = 1.0f / M[(size_t)k * n + k];
    __syncthreads();
    if (tid < n4) {
      const float pv = pivinv;
      float4 mv = ((float4*)(M + (size_t)k * n))[tid];
      float4 rv = ((float4*)(R + (size_t)k * n))[tid];
      mv.x *= pv; mv.y *= pv; mv.z *= pv; mv.w *= pv;
      rv.x *= pv; rv.y *= pv; rv.z *= pv; rv.w *= pv;
      ((float4*)mk)[tid] = mv; ((float4*)(M + (size_t)k * n))[tid] = mv;
      ((float4*)rk)[tid] = rv; ((float4*)(R + (size_t)k * n))[tid] = rv;
    }
    __syncthreads();
    const int i = tid;  // blockDim == n
    if (i != k) {
      const float f = M[(size_t)i * n + k];
      float4* Mr = (float4*)(M + (size_t)i * n);
      float4* Rr = (float4*)(R + (size_t)i * n);
      const float4* mk4 = (const float4*)mk;
      const float4* rk4 = (const float4*)rk;
      for (int j = 0; j < n4; ++j) {
        float4 mv = Mr[j], pa = mk4[j];
        mv.x = fmaf(-f, pa.x, mv.x); mv.y = fmaf(-f, pa.y, mv.y);
        mv.z = fmaf(-f, pa.z, mv.z); mv.w = fmaf(-f, pa.w, mv.w);
        Mr[j] = mv;
        float4 rv = Rr[j], pb = rk4[j];
        rv.x = fmaf(-f, pb.x, rv.x); rv.y = fmaf(-f, pb.y, rv.y);
        rv.z = fmaf(-f, pb.z, rv.z); rv.w = fmaf(-f, pb.w, rv.w);
        Rr[j] = rv;
      }
    }
    __syncthreads();
  }
}

// ---------------------------------------------------------------------------
// Small kernels
// ---------------------------------------------------------------------------
__global__ void vec_init(const float* __restrict__ inputs,
                         const float* __restrict__ input_w,
                         const float* __restrict__ input_b,
                         const float* __restrict__ attn_w,
                         const float* __restrict__ attn_b,
                         float* __restrict__ x, float* __restrict__ s, int n) {
  int i = blockIdx.x * blockDim.x + threadIdx.x;
  if (i < n) {
    float xv = inputs[i] * input_w[i] + input_b[i];
    x[i] = xv;
    s[i] = xv * (attn_w[i] + attn_b[0]);
  }
}

__global__ __launch_bounds__(1024) void max_exp(float* __restrict__ s, int n) {
  __shared__ float sm[1024];
  const int tid = threadIdx.x;
  float m = -INFINITY;
  for (int j = tid; j < n; j += 1024) m = fmaxf(m, s[j]);
  sm[tid] = m;
  __syncthreads();
  for (int off = 512; off > 0; off >>= 1) {
    if (tid < off) sm[tid] = fmaxf(sm[tid], sm[tid + off]);
    __syncthreads();
  }
  const float mx = sm[0];
  for (int j = tid; j < n; j += 1024) s[j] = expf(s[j] - mx);  // s -> att_e
}

__global__ void set_diag_matrix(float* __restrict__ P,
                                const float* __restrict__ d, int n) {
  int idx = blockIdx.x * blockDim.x + threadIdx.x;
  if (idx < n * n) {
    int i = idx / n, j = idx - i * n;
    P[idx] = (i == j) ? d[i] : 0.0f;
  }
}

__global__ void add_diag(float* __restrict__ M, const float* __restrict__ d,
                         int n) {
  int i = blockIdx.x * blockDim.x + threadIdx.x;
  if (i < n) M[(size_t)i * n + i] += d[i];
}

// c[i] = control[i] - mean_j(K[i,j] * x[j]); 256 thr x float4 covers row
__global__ __launch_bounds__(256) void kx_rowmean(
    const float* __restrict__ K, const float* __restrict__ x,
    const float* __restrict__ control, float* __restrict__ c, int n) {
  __shared__ float sm[256];
  const int i = blockIdx.x, tid = threadIdx.x;
  const float4 kv = ((const float4*)(K + (size_t)i * n))[tid];
  const float4 xv = ((const float4*)x)[tid];
  sm[tid] = kv.x * xv.x + kv.y * xv.y + kv.z * xv.z + kv.w * xv.w;
  __syncthreads();
  for (int off = 128; off > 0; off >>= 1) {
    if (tid < off) sm[tid] += sm[tid + off];
    __syncthreads();
  }
  if (tid == 0) c[i] = control[i] - sm[0] / (float)n;
}

// ctrl[j] = sum_i c[i] * lap[i,j]   (coalesced over j)
__global__ __launch_bounds__(256) void vecmat(const float* __restrict__ c,
                                              const float* __restrict__ lap,
                                              float* __restrict__ ctrl, int n) {
  int j = blockIdx.x * blockDim.x + threadIdx.x;
  if (j < n) {
    float s = 0.0f;
    for (int i = 0; i < n; ++i) s = fmaf(c[i], lap[(size_t)i * n + j], s);
    ctrl[j] = s;
  }
}

__device__ __forceinline__ float sigmoidf(float z) {
  return 1.0f / (1.0f + expf(-z));
}

// sensory row sums (float4 over the row: 256 threads x 4 = 1024 cols)
__global__ __launch_bounds__(256) void sensory_rows(
    const float* __restrict__ sw, const float* __restrict__ ssig,
    const float* __restrict__ smu, const float* __restrict__ serev,
    const float* __restrict__ smask, const float* __restrict__ x,
    float* __restrict__ wnum_s, float* __restrict__ wden_s, int n) {
  __shared__ float snum[256];
  __shared__ float sden[256];
  const int i = blockIdx.x, tid = threadIdx.x;
  const size_t ro = (size_t)i * n;
  const float xi = x[i];
  const float4 wv = ((const float4*)(sw + ro))[tid];
  const float4 gv = ((const float4*)(ssig + ro))[tid];
  const float4 mv = ((const float4*)(smu + ro))[tid];
  const float4 ev = ((const float4*)(serev + ro))[tid];
  const float4 kv = ((const float4*)(smask + ro))[tid];
  float a0 = wv.x * sigmoidf(gv.x * (xi - mv.x)) * kv.x;
  float a1 = wv.y * sigmoidf(gv.y * (xi - mv.y)) * kv.y;
  float a2 = wv.z * sigmoidf(gv.z * (xi - mv.z)) * kv.z;
  float a3 = wv.w * sigmoidf(gv.w * (xi - mv.w)) * kv.w;
  snum[tid] = a0 * ev.x + a1 * ev.y + a2 * ev.z + a3 * ev.w;
  sden[tid] = a0 + a1 + a2 + a3;
  __syncthreads();
  for (int off = 128; off > 0; off >>= 1) {
    if (tid < off) { snum[tid] += snum[tid + off]; sden[tid] += sden[tid + off]; }
    __syncthreads();
  }
  if (tid == 0) { wnum_s[i] = snum[0]; wden_s[i] = sden[0]; }
}

// one GCN step: hn[i] = relu(gcw * sum_j adj[i,j]*h[j] + gcb)
__global__ __launch_bounds__(256) void gcn_matvec(
    const float* __restrict__ adj, const float* __restrict__ h,
    const float* __restrict__ gcw, const float* __restrict__ gcb,
    float* __restrict__ hn, int n) {
  __shared__ float sm[256];
  const int i = blockIdx.x, tid = threadIdx.x;
  const float4 av = ((const float4*)(adj + (size_t)i * n))[tid];
  const float4 hv = ((const float4*)h)[tid];
  sm[tid] = av.x * hv.x + av.y * hv.y + av.z * hv.z + av.w * hv.w;
  __syncthreads();
  for (int off = 128; off > 0; off >>= 1) {
    if (tid < off) sm[tid] += sm[tid + off];
    __syncthreads();
  }
  if (tid == 0) hn[i] = fmaxf(gcw[0] * sm[0] + gcb[0], 0.0f);
}

__global__ void cmt_init(const float* __restrict__ d,
                         const float* __restrict__ cm,
                         const float* __restrict__ ctrl,
                         float* __restrict__ cm_t, float* __restrict__ v,
                         int n) {
  int i = blockIdx.x * blockDim.x + threadIdx.x;
  if (i < n) {
    cm_t[i] = d[i] + cm[i] * (float)ODE_UNFOLDS + ctrl[i];
    v[i] = 0.0f;
  }
}

// one ODE unfold: float4 row reduction + in-place v[i] update
__global__ __launch_bounds__(256) void ode_step(
    const float* __restrict__ w, const float* __restrict__ sig,
    const float* __restrict__ mu, const float* __restrict__ erev,
    const float* __restrict__ mask, const float* __restrict__ gleak,
    const float* __restrict__ vleak, const float* __restrict__ cm_t,
    const float* __restrict__ wnum_s, const float* __restrict__ wden_s,
    float* __restrict__ v, int n) {
  __shared__ float snum[256];
  __shared__ float sden[256];
  const int i = blockIdx.x, tid = threadIdx.x;
  const size_t ro = (size_t)i * n;
  const float vi = v[i];
  const float4 wv = ((const float4*)(w + ro))[tid];
  const float4 gv = ((const float4*)(sig + ro))[tid];
  const float4 mv = ((const float4*)(mu + ro))[tid];
  const float4 ev = ((const float4*)(erev + ro))[tid];
  const float4 kv = ((const float4*)(mask + ro))[tid];
  float a0 = wv.x * sigmoidf(gv.x * (vi - mv.x)) * kv.x;
  float a1 = wv.y * sigmoidf(gv.y * (vi - mv.y)) * kv.y;
  float a2 = wv.z * sigmoidf(gv.z * (vi - mv.z)) * kv.z;
  float a3 = wv.w * sigmoidf(gv.w * (vi - mv.w)) * kv.w;
  snum[tid] = a0 * ev.x + a1 * ev.y + a2 * ev.z + a3 * ev.w;
  sden[tid] = a0 + a1 + a2 + a3;
  __syncthreads();
  for (int off = 128; off > 0; off >>= 1) {
    if (tid < off) { snum[tid] += snum[tid + off]; sden[tid] += sden[tid + off]; }
    __syncthreads();
  }
  if (tid == 0) {
    float wn = snum[0] + wnum_s[i];
    float wd = sden[0] + wden_s[i];
    v[i] = (cm_t[i] * vi + gleak[i] * vleak[i] + wn) /
           (cm_t[i] + gleak[i] + wd + EPSV);
  }
}

__global__ void finalize(const float* __restrict__ v,
                         const float* __restrict__ output_w,
                         const float* __restrict__ output_b,
                         const float* __restrict__ ctrl,
                         const float* __restrict__ control,
                         float* __restrict__ out, int n) {
  int i = blockIdx.x * blockDim.x + threadIdx.x;
  if (i < n) {
    out[i] = v[i] * output_w[i] + output_b[i];              // out
    out[n + i] = v[i];                                      // v
    out[2 * n + n * n + i] = ctrl[i] + control[i];          // ctrl + control
  }
}

// ---------------------------------------------------------------------------
extern "C" void kernel_launch(void* const* d_in, const int* in_sizes, int n_in,
                              void* d_out, int out_size, void* d_ws,
                              size_t ws_size, hipStream_t stream) {
  const int n = NN;
  const float* inputs   = (const float*)d_in[0];
  const float* adj      = (const float*)d_in[1];
  const float* lap      = (const float*)d_in[2];
  const float* control  = (const float*)d_in[3];
  const float* gleak    = (const float*)d_in[4];
  const float* vleak    = (const float*)d_in[5];
  const float* cm       = (const float*)d_in[6];
  const float* sigma    = (const float*)d_in[7];
  const float* mu       = (const float*)d_in[8];
  const float* w        = (const float*)d_in[9];
  const float* erev     = (const float*)d_in[10];
  const float* ssig     = (const float*)d_in[11];
  const float* smu      = (const float*)d_in[12];
  const float* sw       = (const float*)d_in[13];
  const float* serev    = (const float*)d_in[14];
  const float* mask     = (const float*)d_in[15];
  const float* smask    = (const float*)d_in[16];
  const float* input_w  = (const float*)d_in[17];
  const float* input_b  = (const float*)d_in[18];
  const float* output_w = (const float*)d_in[19];
  const float* output_b = (const float*)d_in[20];
  const float* attn_w   = (const float*)d_in[21];
  const float* attn_b   = (const float*)d_in[22];
  const float* gc_w     = (const float*)d_in[23];
  const float* gc_b     = (const float*)d_in[24];

  float* out = (float*)d_out;

  // workspace layout: 6 NxN matrices + vectors (~24.2 MB)
  const size_t nn = (size_t)n * n;
  float* W    = (float*)d_ws;
  float* P    = W;
  float* T1   = W + 1 * nn;
  float* M    = W + 2 * nn;
  float* K    = W + 3 * nn;   // RHS, becomes K after solve
  float* T3   = W + 4 * nn;
  float* T4   = W + 5 * nn;
  float* base = W + 6 * nn;
  float* x      = base + 0 * n;
  float* att_e  = base + 1 * n;   // holds s then exp(s - max)
  float* ctrl   = base + 2 * n;
  float* cvec   = base + 3 * n;
  float* wnum_s = base + 4 * n;
  float* wden_s = base + 5 * n;
  float* hA     = base + 6 * n;
  float* hB     = base + 7 * n;
  float* cm_t   = base + 8 * n;
  float* v      = base + 9 * n;

  const dim3 gemm_grid(n / 64, n / 64);
  const dim3 gemm_blk(128);
  const int vb = 256, vg = (n + vb - 1) / vb;

  // --- attention / Q,R ---
  vec_init<<<vg, vb, 0, stream>>>(inputs, input_w, input_b, attn_w, attn_b,
                                  x, att_e, n);
  max_exp<<<1, 1024, 0, stream>>>(att_e, n);
  set_diag_matrix<<<(int)((nn + 255) / 256), 256, 0, stream>>>(P, att_e, n);

  // --- Riccati iterations (A=adj, B=lap, Q=R=diag(att_e)) ---
  for (int it = 0; it < RICCATI_ITERS; ++it) {
    // T1 = B^T P
    gemm_f32_wmma<1><<<gemm_grid, gemm_blk, 0, stream>>>(lap, P, T1, 1.f, 0.f, n);
    // M = T1 B + R
    gemm_f32_wmma<0><<<gemm_grid, gemm_blk, 0, stream>>>(T1, lap, M, 1.f, 0.f, n);
    add_diag<<<vg, vb, 0, stream>>>(M, att_e, n);
    // K = M^-1 (T1 A)
    gemm_f32_wmma<0><<<gemm_grid, gemm_blk, 0, stream>>>(T1, adj, K, 1.f, 0.f, n);
    gauss_jordan_solve<<<1, 1024, 0, stream>>>(M, K, n);
    // T3 = A - B K
    hipMemcpyAsync(T3, adj, nn * sizeof(float), hipMemcpyDeviceToDevice, stream);
    gemm_f32_wmma<0><<<gemm_grid, gemm_blk, 0, stream>>>(lap, K, T3, -1.f, 1.f, n);
    // P = A^T P T3 + Q
    gemm_f32_wmma<1><<<gemm_grid, gemm_blk, 0, stream>>>(adj, P, T4, 1.f, 0.f, n);
    gemm_f32_wmma<0><<<gemm_grid, gemm_blk, 0, stream>>>(T4, T3, P, 1.f, 0.f, n);
    add_diag<<<vg, vb, 0, stream>>>(P, att_e, n);
  }
  // final K
  gemm_f32_wmma<1><<<gemm_grid, gemm_blk, 0, stream>>>(lap, P, T1, 1.f, 0.f, n);
  gemm_f32_wmma<0><<<gemm_grid, gemm_blk, 0, stream>>>(T1, lap, M, 1.f, 0.f, n);
  add_diag<<<vg, vb, 0, stream>>>(M, att_e, n);
  gemm_f32_wmma<0><<<gemm_grid, gemm_blk, 0, stream>>>(T1, adj, K, 1.f, 0.f, n);
  gauss_jordan_solve<<<1, 1024, 0, stream>>>(M, K, n);

  // --- ctrl = (control - mean(K*x, axis=1)) @ lap ---
  kx_rowmean<<<n, 256, 0, stream>>>(K, x, control, cvec, n);
  vecmat<<<vg, vb, 0, stream>>>(cvec, lap, ctrl, n);

  // --- sensory row sums ---
  sensory_rows<<<n, 256, 0, stream>>>(sw, ssig, smu, serev, smask, x,
                                      wnum_s, wden_s, n);

  // --- 4 GCN steps: x -> hA -> hB -> hA -> hB ---
  gcn_matvec<<<n, 256, 0, stream>>>(adj, x,  gc_w, gc_b, hA, n);
  gcn_matvec<<<n, 256, 0, stream>>>(adj, hA, gc_w, gc_b, hB, n);
  gcn_matvec<<<n, 256, 0, stream>>>(adj, hB, gc_w, gc_b, hA, n);
  gcn_matvec<<<n, 256, 0, stream>>>(adj, hA, gc_w, gc_b, hB, n);

  cmt_init<<<vg, vb, 0, stream>>>(hB, cm, ctrl, cm_t, v, n);

  // --- ODE unfolds ---
  for (int u = 0; u < ODE_UNFOLDS; ++u) {
    ode_step<<<n, 256, 0, stream>>>(w, sigma, mu, erev, mask, gleak, vleak,
                                    cm_t, wnum_s, wden_s, v, n);
  }

  // --- outputs: (out, v, laplacian, ctrl + control) ---
  hipMemcpyAsync(out + 2 * n, lap, nn * sizeof(float),
                 hipMemcpyDeviceToDevice, stream);
  finalize<<<vg, vb, 0, stream>>>(v, output_w, output_b, ctrl, control, out, n);
}